// SimpleUnsharedPatchScorer_83562883711078
// MI455X (gfx1250) — compile-verified
//
#include <hip/hip_runtime.h>
#include <hip/hip_bf16.h>

// SimpleUnsharedPatchScorer for MI455X (gfx1250, wave32).
//
// scores[b,p] = sum_d x[scramble(p*768+d)] * W[p,d] + bias[p]
// B=256, P_total=196, D=768.  HBM-bound (0.5 FLOP/byte); x (154MB) fits in
// the 192MB L2, so the scrambled gather costs no extra HBM traffic.
//
// WMMA mapping: V_WMMA_F32_16X16X4_F32 with M=16 batches, K=4 d-chunk,
// N=16 replicated weight columns (D replicated over N; 16x FLOP waste is
// free at this arithmetic intensity).
//
// f32 operand layout (ISA 7.12.2): A 16x4 -> K = vgpr + 2*(lane>=16).
// B 4x16 assumed symmetric -> same K = vgpr + 2*lanehalf. Each lane feeds
// the WMMA the pair (d0, d0+1) with d0 = kc + 2*lanehalf for BOTH A and B.
//
// Addressing: x is 38.6M elements (<2^31), so per-lane indices are kept as
// unsigned 32-bit and the compiler can use the SADDR + 32-bit VGPR-offset
// global addressing mode (1 VALU add per gather instead of a 64-bit chain).

typedef __attribute__((ext_vector_type(2))) float    v2f;
typedef __attribute__((ext_vector_type(8))) float    v8f;
typedef __attribute__((ext_vector_type(2))) unsigned v2u;

#define NPATCH   196
#define DDIM     768
#define XB       150528u   // 3*224*224 floats per batch image
#define CSTRIDE  50176u    // 224*224

__global__ __launch_bounds__(128) void patch_scorer_wmma(
    const float* __restrict__ x,
    const float* __restrict__ weight,
    const float* __restrict__ bias,
    float* __restrict__ out)
{
    __shared__ __align__(16) unsigned s_off[DDIM];  // x element offsets for this patch
    __shared__ __align__(16) float    s_w[DDIM];    // weight row for this patch

    const unsigned p = blockIdx.x;                  // patch index 0..195

    // ---- Phase 1: precompute scrambled offsets + stage weights in LDS ----
    // L = p*768 + d over dims (pi,pj,c,i,j) = (16,16,3,14,14)
    for (int d = threadIdx.x; d < DDIM; d += 128) {
        unsigned L    = p * (unsigned)DDIM + (unsigned)d;
        unsigned pipj = L / 588u;
        unsigned rem  = L - pipj * 588u;
        unsigned c    = rem / 196u;
        unsigned r2   = rem - c * 196u;
        unsigned i    = r2 / 14u;
        unsigned j    = r2 - i * 14u;
        unsigned pi   = pipj >> 4;
        unsigned pj   = pipj & 15u;
        s_off[d] = c * CSTRIDE + (i * 16u + pi) * 224u + (j * 16u + pj);
        s_w[d]   = weight[(size_t)p * DDIM + d];
    }
    __syncthreads();

    // ---- Phase 2: per-wave 16-batch tile, 192 accumulating f32 WMMAs ----
    const unsigned lane = threadIdx.x & 31u;
    const unsigned wv   = threadIdx.x >> 5;         // wave in block: 0..3
    const unsigned m    = lane & 15u;               // batch row within tile (A: M=lane%16)
    const unsigned hh   = lane >> 4;                // lane half selects K pair
    const unsigned b0   = blockIdx.y * 64u + wv * 16u;

    const unsigned idxBase = (b0 + m) * XB;         // 32-bit per-lane element base

    v8f acc = {0.f, 0.f, 0.f, 0.f, 0.f, 0.f, 0.f, 0.f};

    #pragma unroll 4
    for (int kc = 0; kc < DDIM; kc += 4) {
        const int d0 = kc + (int)(hh << 1);         // this lane's K pair: d0, d0+1
        const v2u off = *(const v2u*)&s_off[d0];    // ds_load_b64 (8B aligned: d0 even)
        v2f a;
        a.x = x[idxBase + off.x];                   // saddr + 32-bit voffset gathers
        a.y = x[idxBase + off.y];
        const v2f b = *(const v2f*)&s_w[d0];        // ds_load_b64 weight pair

        // v_wmma_f32_16x16x4_f32: D = A(16x4) * B(4x16) + C
        acc = __builtin_amdgcn_wmma_f32_16x16x4_f32(
            false, a, false, b, (short)0, acc, false, false);
    }

    // ---- Epilogue: C/D layout -> lanes 0/16 hold M=0..7 / M=8..15 in acc[0..7]
    const float bp = bias[p];
    if (m == 0u) {
        const unsigned mb = hh * 8u;
        #pragma unroll
        for (int v = 0; v < 8; ++v)
            out[(size_t)(b0 + mb + (unsigned)v) * NPATCH + p] = acc[v] + bp;
    }
}

extern "C" void kernel_launch(void* const* d_in, const int* in_sizes, int n_in,
                              void* d_out, int out_size, void* d_ws, size_t ws_size,
                              hipStream_t stream) {
    const float* x      = (const float*)d_in[0];   // [B,3,224,224]
    const float* weight = (const float*)d_in[1];   // [196,768]
    const float* bias   = (const float*)d_in[2];   // [196]
    float*       out    = (float*)d_out;           // [B,196]

    const int B = in_sizes[0] / (int)XB;           // 256
    dim3 grid(NPATCH, (unsigned)(B / 64), 1);      // patch x 64-batch groups
    dim3 block(128, 1, 1);                         // 4 waves: one 16-batch tile each
    patch_scorer_wmma<<<grid, block, 0, stream>>>(x, weight, bias, out);
}